// YoloLayer_81879256531616
// MI455X (gfx1250) — compile-verified
//
#include <hip/hip_runtime.h>
#include <math.h>

// ---------------- problem constants (fixed by reference setup) ----------------
#define NB 16
#define NT 50
#define NA 3
#define NH 76
#define NW 76
#define NC 80
#define ATT 85                      // 5 + NUM_CLASS
#define NCH (NA * ATT)              // 255
#define CPM (NH * NW)               // 5776 cells per (b, a) map
#define CPB (NA * CPM)              // 17328 cells per batch
#define K2_BLOCK 256
#define K2_BPB 68                   // ceil(17328 / 256)
#define EPSL 1e-12f
#define NSTG (6 * NT)               // staged floats per batch: x1|x2|y1|y2|area|cellid

// workspace layout (floats)
#define NREC (NB * NT)              // 800
enum {
    OFF_GX = 0,          OFF_GY = NREC,       OFF_GW = 2 * NREC,   OFF_GH = 3 * NREC,
    OFF_VALID = 4 * NREC, OFF_WRITE = 5 * NREC, OFF_ALOC = 6 * NREC, OFF_GI = 7 * NREC,
    OFF_GJ = 8 * NREC,   OFF_TX = 9 * NREC,   OFF_TY = 10 * NREC,  OFF_TW = 11 * NREC,
    OFF_TH = 12 * NREC,  OFF_COORD = 13 * NREC, OFF_CLSI = 14 * NREC, OFF_ISLAST = 15 * NREC,
    OFF_PART = 16 * NREC
};
#define NPART2 (NB * K2_BPB)        // 1088 partials from the streaming pass
#define NPART (NPART2 + NREC)       // + 800 partials from the obj-cell pass = 1888
#define OFF_STAGE (OFF_PART + NPART)
#define NCHUNK ((NPART + 63) / 64)  // 64 partials per WMMA step

// gfx1250 async-to-LDS path (ASYNCcnt), guarded so both toolchains compile
#if __has_builtin(__builtin_amdgcn_global_load_async_to_lds_b32) && \
    __has_builtin(__builtin_amdgcn_s_wait_asynccnt)
#define USE_ASYNC_LDS 1
// signature (from hipcc diagnostic): param0 = AS(1) int*, param1 = AS(3) int*, imm offset, imm cpol
typedef __attribute__((address_space(1))) int async_gptr;
typedef __attribute__((address_space(3))) int async_lptr;
#else
#define USE_ASYNC_LDS 0
#endif

__device__ const float ANCW9[9] = {10.f, 16.f, 33.f, 30.f, 62.f, 59.f, 116.f, 156.f, 373.f};
__device__ const float ANCH9[9] = {13.f, 30.f, 23.f, 61.f, 45.f, 119.f, 90.f, 198.f, 326.f};

__device__ __forceinline__ float sig(float v) { return 1.0f / (1.0f + expf(-v)); }

// ---------------- kernel 1: build per-target records + staged IoU table ----------------
__global__ void k1_targets(const float* __restrict__ tgt, const int* __restrict__ imgdim,
                           float* __restrict__ ws) {
    int b = threadIdx.x;
    if (b >= NB) return;
    float stride = (float)imgdim[0] / (float)NH;
    float aw[9], ah[9];
    for (int k = 0; k < 9; ++k) { aw[k] = ANCW9[k] / stride; ah[k] = ANCH9[k] / stride; }

    bool vrun = true;
    float* stage = ws + OFF_STAGE + b * NSTG;
    for (int t = 0; t < NT; ++t) {
        const float* r = tgt + ((size_t)b * NT + t) * 5;
        float t0 = r[0], t1 = r[1], t2 = r[2], t3 = r[3], t4 = r[4];
        vrun = vrun && ((t0 + t1 + t2 + t3 + t4) != 0.0f);   // cumprod of nonzero rows
        float gx = t1 * NW, gy = t2 * NH, gw = t3 * NW, gh = t4 * NH;
        int gi = (int)gx; gi = gi < 0 ? 0 : (gi > NW - 1 ? NW - 1 : gi);
        int gj = (int)gy; gj = gj < 0 ? 0 : (gj > NH - 1 ? NH - 1 : gj);
        // best of 9 anchors by wh-IoU (first max wins, matching jnp.argmax)
        int best = 0; float bestv = -1.0f;
        for (int k = 0; k < 9; ++k) {
            float inter = fminf(gw, aw[k]) * fminf(gh, ah[k]);
            float uni = gw * gh + aw[k] * ah[k] - inter;
            float ratio = inter / (uni + 1e-16f);
            if (ratio > bestv) { bestv = ratio; best = k; }
        }
        int aloc = (best < NA) ? best : -1;                  // lut over ANCHORS_MASK=[0,1,2]
        bool wr = vrun && (aloc >= 0);
        int alc = aloc < 0 ? 0 : aloc;
        int clsi = (int)t0; clsi = clsi < 0 ? 0 : (clsi > NC - 1 ? NC - 1 : clsi);

        int idx = b * NT + t;
        ws[OFF_GX + idx] = gx;  ws[OFF_GY + idx] = gy;
        ws[OFF_GW + idx] = gw;  ws[OFF_GH + idx] = gh;
        ws[OFF_VALID + idx] = vrun ? 1.f : 0.f;
        ws[OFF_WRITE + idx] = wr ? 1.f : 0.f;
        ws[OFF_ALOC + idx] = (float)alc;
        ws[OFF_GI + idx] = (float)gi;  ws[OFF_GJ + idx] = (float)gj;
        ws[OFF_TX + idx] = gx - (float)gi;  ws[OFF_TY + idx] = gy - (float)gj;
        ws[OFF_TW + idx] = logf(gw / aw[alc] + 1e-16f);
        ws[OFF_TH + idx] = logf(gh / ah[alc] + 1e-16f);
        ws[OFF_COORD + idx] = sqrtf(2.0f - t3 * t4);
        ws[OFF_CLSI + idx] = (float)clsi;

        // staged table for the streaming pass: degenerate boxes kill invalid targets
        // (inter==0 -> never ignored), write-cell id packed as an exact float (-1 if none)
        if (vrun) {
            stage[0 * NT + t] = gx - gw * 0.5f;
            stage[1 * NT + t] = gx + gw * 0.5f;
            stage[2 * NT + t] = gy - gh * 0.5f;
            stage[3 * NT + t] = gy + gh * 0.5f;
            stage[4 * NT + t] = gw * gh;
        } else {
            stage[0 * NT + t] = 1e30f;
            stage[1 * NT + t] = -1e30f;
            stage[2 * NT + t] = 1e30f;
            stage[3 * NT + t] = -1e30f;
            stage[4 * NT + t] = 0.0f;
        }
        stage[5 * NT + t] = wr ? (float)(alc * CPM + gj * NW + gi) : -1.0f;
    }
    // last-wins resolution for duplicate scatter cells (matches .at[].set semantics)
    for (int t = 0; t < NT; ++t) {
        int idx = b * NT + t;
        float last = ws[OFF_WRITE + idx];
        if (last != 0.f) {
            int a = (int)ws[OFF_ALOC + idx], gi = (int)ws[OFF_GI + idx], gj = (int)ws[OFF_GJ + idx];
            for (int u = t + 1; u < NT; ++u) {
                int j2 = b * NT + u;
                if (ws[OFF_WRITE + j2] != 0.f && (int)ws[OFF_ALOC + j2] == a &&
                    (int)ws[OFF_GI + j2] == gi && (int)ws[OFF_GJ + j2] == gj) { last = 0.f; break; }
            }
        }
        ws[OFF_ISLAST + idx] = last;
    }
}

// ---------------- kernel 2: dense streaming pass (conf BCE + ignore IoUs) ----------------
__global__ void k2_stream(const float* __restrict__ X, const int* __restrict__ imgdim,
                          const float* __restrict__ ws, float* __restrict__ partials) {
    __shared__ float stg[NSTG];
    __shared__ float red[K2_BLOCK];

    int b = blockIdx.y;
    int tid = threadIdx.x;
    const float* gstage = ws + OFF_STAGE + b * NSTG;
#if USE_ASYNC_LDS
    for (int k = tid; k < NSTG; k += K2_BLOCK)
        __builtin_amdgcn_global_load_async_to_lds_b32(
            (async_gptr*)(gstage + k), (async_lptr*)(&stg[k]), 0, 0);
    __builtin_amdgcn_s_wait_asynccnt(0);
#else
    for (int k = tid; k < NSTG; k += K2_BLOCK) stg[k] = gstage[k];
#endif
    __syncthreads();

    float stride = (float)imgdim[0] / (float)NH;
    int cell = blockIdx.x * K2_BLOCK + tid;
    float term = 0.0f;
    if (cell < CPB) {
        int a = cell / CPM;
        int rem = cell - a * CPM;
        int j = rem / NW, i = rem - j * NW;
        const float* px = X + ((size_t)b * NCH + (size_t)a * ATT) * CPM + rem;
        float x0 = px[0], x1 = px[CPM], x2 = px[2 * CPM], x3 = px[3 * CPM], x4 = px[4 * CPM];
        float pc = sig(x4);
        float bw = expf(x2) * (ANCW9[a] / stride);
        float bh = expf(x3) * (ANCH9[a] / stride);
        float bx = sig(x0) + (float)i, by = sig(x1) + (float)j;
        float bx1 = bx - bw * 0.5f, bx2 = bx + bw * 0.5f;
        float by1 = by - bh * 0.5f, by2 = by + bh * 0.5f;
        float barea = bw * bh;
        float cellf = (float)cell;
        bool ign = false, obj = false;
        #pragma unroll 5
        for (int t = 0; t < NT; ++t) {
            float iw = fmaxf(fminf(bx2, stg[1 * NT + t]) - fmaxf(bx1, stg[0 * NT + t]), 0.f);
            float ih = fmaxf(fminf(by2, stg[3 * NT + t]) - fmaxf(by1, stg[2 * NT + t]), 0.f);
            float inter = iw * ih;
            // inter/(union+1e-16) > 0.7  <=>  inter > 0.7*(union+1e-16)   (union > 0)
            ign = ign || (inter > 0.7f * (barea + stg[4 * NT + t] - inter + 1e-16f));
            obj = obj || (stg[5 * NT + t] == cellf);
        }
        float noobj = ign ? 0.f : 1.f;
        // obj-part BCE: t=obj, p=pc*obj ; noobj-part BCE: t=0, p=pc*noobj
        float objterm = obj ? logf(pc + EPSL) : logf(1.0f + EPSL);
        term = -(objterm + logf(1.0f - pc * noobj + EPSL));
    }
    red[tid] = term;
    __syncthreads();
    for (int s = K2_BLOCK / 2; s > 0; s >>= 1) {
        if (tid < s) red[tid] += red[tid + s];
        __syncthreads();
    }
    if (tid == 0) partials[b * K2_BPB + blockIdx.x] = red[0];
}

// ---------------- kernel 3: sparse obj-cell pass (coord MSE + 80-class BCE) ----------------
__global__ void k3_obj(const float* __restrict__ X, const float* __restrict__ ws,
                       float* __restrict__ partials) {
    int b = blockIdx.x;
    int t = threadIdx.x;
    if (t >= NT) return;
    int idx = b * NT + t;
    float s = 0.f;
    if (ws[OFF_ISLAST + idx] != 0.f) {
        int a = (int)ws[OFF_ALOC + idx], gi = (int)ws[OFF_GI + idx], gj = (int)ws[OFF_GJ + idx];
        // tcls is a set() scatter with the class in the index -> union of classes at this cell
        unsigned long long m0 = 0ull, m1 = 0ull;
        for (int u = 0; u < NT; ++u) {
            int j2 = b * NT + u;
            if (ws[OFF_WRITE + j2] != 0.f && (int)ws[OFF_ALOC + j2] == a &&
                (int)ws[OFF_GI + j2] == gi && (int)ws[OFF_GJ + j2] == gj) {
                int c = (int)ws[OFF_CLSI + j2];
                if (c < 64) m0 |= 1ull << c; else m1 |= 1ull << (c - 64);
            }
        }
        const float* px = X + ((size_t)b * NCH + (size_t)a * ATT) * CPM + (size_t)gj * NW + gi;
        float p0 = sig(px[0]);
        float p1 = sig(px[CPM]);
        float pw = px[2 * CPM], ph = px[3 * CPM];
        float cm = ws[OFF_COORD + idx];
        float dx = p0 - ws[OFF_TX + idx], dy = p1 - ws[OFF_TY + idx];
        float dw = pw - ws[OFF_TW + idx], dh = ph - ws[OFF_TH + idx];
        s = 0.5f * cm * cm * (dx * dx + dy * dy + dw * dw + dh * dh);
        for (int c = 0; c < NC; ++c) {
            float pcl = sig(px[(size_t)(5 + c) * CPM]);
            bool tc = (c < 64) ? ((m0 >> c) & 1ull) : ((m1 >> (c - 64)) & 1ull);
            s += tc ? -logf(pcl + EPSL) : -logf(1.0f - pcl + EPSL);
        }
    }
    partials[NPART2 + idx] = s;
}

// ---------------- kernel 4: deterministic WMMA-based final reduction ----------------
// A (16x4 f32) holds 64 partials striped over the wave; B = ones(4x16).
// D[m][n] = sum_k A[m][k] accumulated into C across chunks; total = sum_m D[m][0],
// i.e. (v0+..+v7) on lane 0 (M=0..7) plus the same on lane 16 (M=8..15).
typedef __attribute__((ext_vector_type(2))) float v2f;
typedef __attribute__((ext_vector_type(8))) float v8f;

__global__ void k4_reduce(const float* __restrict__ partials, float* __restrict__ out) {
    int lane = threadIdx.x;          // exactly one wave32, EXEC all-ones around WMMA
    v8f acc = {};
    v2f ones = {1.0f, 1.0f};
    for (int ch = 0; ch < NCHUNK; ++ch) {
        int i0 = ch * 64 + lane;
        int i1 = i0 + 32;
        v2f a;
        a.x = (i0 < NPART) ? partials[i0] : 0.0f;
        a.y = (i1 < NPART) ? partials[i1] : 0.0f;
        acc = __builtin_amdgcn_wmma_f32_16x16x4_f32(
            /*neg_a=*/false, a, /*neg_b=*/false, ones,
            /*c_mod=*/(short)0, acc, /*reuse_a=*/false, /*reuse_b=*/false);
    }
    float s = acc[0] + acc[1] + acc[2] + acc[3] + acc[4] + acc[5] + acc[6] + acc[7];
    float total = __shfl(s, 0, 32) + __shfl(s, 16, 32);
    if (lane == 0) out[0] = total;
}

// ---------------- launcher ----------------
extern "C" void kernel_launch(void* const* d_in, const int* in_sizes, int n_in,
                              void* d_out, int out_size, void* d_ws, size_t ws_size,
                              hipStream_t stream) {
    const float* x = (const float*)d_in[0];
    const float* target = (const float*)d_in[1];
    const int* imgdim = (const int*)d_in[2];
    float* ws = (float*)d_ws;
    float* partials = ws + OFF_PART;

    k1_targets<<<1, NB, 0, stream>>>(target, imgdim, ws);
    dim3 g2(K2_BPB, NB, 1);
    k2_stream<<<g2, K2_BLOCK, 0, stream>>>(x, imgdim, ws, partials);
    k3_obj<<<NB, 64, 0, stream>>>(x, ws, partials);
    k4_reduce<<<1, 32, 0, stream>>>(partials, (float*)d_out);
}